// DILATELoss_43619687858508
// MI455X (gfx1250) — compile-verified
//
#include <hip/hip_runtime.h>
#include <hip/hip_bf16.h>

typedef __attribute__((ext_vector_type(2))) float v2f;
typedef __attribute__((ext_vector_type(8))) float v8f;

#define NB 128
#define NT 256
#define ND 64
#define GAMMA_F 0.01f
#define BIG_F 1e10f

// LDS layout for the soft-DTW kernel
#define RP    258                 // pitch: stride between threads on a diagonal = 257 words -> conflict-free
#define RROWS 257                 // R is (T+1) x (T+1) logical, stored with pitch RP
#define EBUF  264                 // one rolling E diagonal buffer (>= 257, padded)

// ---------------------------------------------------------------------------
// Kernel 0: row norms ||preds[b,t]||^2 and ||targets[b,s]||^2
// ---------------------------------------------------------------------------
__global__ void dilate_norms(const float* __restrict__ preds,
                             const float* __restrict__ targets,
                             float* __restrict__ pn, float* __restrict__ tn) {
    int idx = blockIdx.x * blockDim.x + threadIdx.x;   // 0 .. NB*NT-1
    if (idx >= NB * NT) return;
    const float* p = preds   + (size_t)idx * ND;
    const float* t = targets + (size_t)idx * ND;
    float sp = 0.f, st = 0.f;
#pragma unroll 8
    for (int k = 0; k < ND; ++k) { sp += p[k] * p[k]; st += t[k] * t[k]; }
    pn[idx] = sp; tn[idx] = st;
}

// ---------------------------------------------------------------------------
// Kernel 1: distance matrix via V_WMMA_F32_16X16X4_F32.
// One wave computes one 16x16 tile of D[b] = pn + tn - 2 * P @ T^T, clamped >= 0.
// A fragment (16x4 f32, 2 VGPRs): lane m=lane&15 holds rows, half=lane>>4 selects K pair.
// B fragment (4x16 f32, 2 VGPRs): lane n=lane&15 holds cols, half selects K pair.
// C/D (16x16 f32, 8 VGPRs): row = r + 8*half, col = lane&15.
// ---------------------------------------------------------------------------
__global__ void dilate_dist_wmma(const float* __restrict__ preds,
                                 const float* __restrict__ targets,
                                 const float* __restrict__ pn,
                                 const float* __restrict__ tn,
                                 float* __restrict__ Dg) {
    int tid   = threadIdx.x;
    int wave  = tid >> 5;
    int lane  = tid & 31;
    int gtile = blockIdx.x * 8 + wave;      // 8 waves / block, 32768 tiles total
    int b     = gtile >> 8;                 // 256 tiles per batch
    int twb   = gtile & 255;
    int tm    = (twb >> 4) << 4;            // tile row base
    int tnb   = (twb & 15) << 4;            // tile col base
    int m     = lane & 15;
    int half  = lane >> 4;

    const float* Pb = preds   + (size_t)b * NT * ND;
    const float* Tb = targets + (size_t)b * NT * ND;

    v8f acc = {};
#pragma unroll
    for (int k0 = 0; k0 < ND; k0 += 4) {
        v2f a, bf;
        a.x  = Pb[(tm  + m) * ND + k0 + 2 * half];
        a.y  = Pb[(tm  + m) * ND + k0 + 2 * half + 1];
        bf.x = Tb[(tnb + m) * ND + k0 + 2 * half];      // B[k][n] = targets[n][k]
        bf.y = Tb[(tnb + m) * ND + k0 + 2 * half + 1];
        acc = __builtin_amdgcn_wmma_f32_16x16x4_f32(
            false, a, false, bf, (short)0, acc, false, false);
    }

    int brow = b * NT;
    float tnv = tn[brow + tnb + m];
#pragma unroll
    for (int r = 0; r < 8; ++r) {
        int row = tm + r + 8 * half;
        int col = tnb + m;
        float v = pn[brow + row] + tnv - 2.0f * acc[r];
        Dg[(size_t)b * NT * NT + row * NT + col] = fmaxf(v, 0.0f);
    }
}

// ---------------------------------------------------------------------------
// Kernel 2: fused soft-DTW forward + backward (gradient) per batch.
// One workgroup per batch; full R matrix lives in LDS (CDNA5: 320 KB/WGP).
// Thread i owns row i; anti-diagonal wavefront with one barrier per diagonal.
// Backward uses 3 rolling E-diagonal buffers; E is reduced on the fly into
// sum(E) and sum(E * omega). Writes per-batch {R_final, sumEw, sumE}.
// ---------------------------------------------------------------------------
__global__ void dilate_softdtw(const float* __restrict__ Dg,
                               float* __restrict__ partials) {
    extern __shared__ float smem[];
    float* R   = smem;                       // RROWS * RP floats
    float* Eb  = smem + RROWS * RP;          // 3 * EBUF floats
    float* red = Eb + 3 * EBUF;              // 256 floats

    const int b = blockIdx.x;
    const int i = threadIdx.x;               // row 0..255
    const float* Db  = Dg + (size_t)b * NT * NT;
    const float invg = 1.0f / GAMMA_F;

    // boundaries: R[0][0] = 0, R[0][j>0] = R[i>0][0] = BIG
    R[i] = (i == 0) ? 0.0f : BIG_F;
    if (i == 0) R[NT] = BIG_F;
    R[(i + 1) * RP] = BIG_F;
    __syncthreads();

    // ---- forward: R[i+1][j+1] = D[i][j] + softmin(diag, up, left) ----
    for (int d = 0; d < 2 * NT - 1; ++d) {
        int j = d - i;
        if (j >= 0 && j < NT) {
            float x  = R[i * RP + j];           // R_{i,j}     (diag)
            float y  = R[i * RP + j + 1];       // R_{i,j+1}   (up)
            float z  = R[(i + 1) * RP + j];     // R_{i+1,j}   (left)
            float mn = fminf(fminf(x, y), z);
            float s  = __expf((mn - x) * invg) + __expf((mn - y) * invg) +
                       __expf((mn - z) * invg);
            float sm = mn - GAMMA_F * __logf(s);
            R[(i + 1) * RP + j + 1] = Db[i * NT + j] + sm;
        }
        __syncthreads();
    }

    // ---- backward: E recursion over reversed anti-diagonals ----
    float sumE = 0.f, sumEw = 0.f;
    const float wscale = 1.0f / (255.0f * 255.0f);
    for (int d = 2 * NT - 2; d >= 0; --d) {
        float* Ec = Eb + (d % 3) * EBUF;          // diagonal d (writing)
        float* E1 = Eb + ((d + 1) % 3) * EBUF;    // diagonal d+1
        float* E2 = Eb + ((d + 2) % 3) * EBUF;    // diagonal d+2
        int j = d - i;
        float e = 0.f;
        if (j >= 0 && j < NT) {
            if (i == NT - 1 && j == NT - 1) {
                e = 1.0f;
            } else {
                float rij = R[(i + 1) * RP + j + 1];
                if (i + 1 < NT)
                    e += __expf((R[(i + 2) * RP + j + 1] - rij -
                                 Db[(i + 1) * NT + j]) * invg) * E1[i + 1];
                if (j + 1 < NT)
                    e += __expf((R[(i + 1) * RP + j + 2] - rij -
                                 Db[i * NT + j + 1]) * invg) * E1[i];
                if (i + 1 < NT && j + 1 < NT)
                    e += __expf((R[(i + 2) * RP + j + 2] - rij -
                                 Db[(i + 1) * NT + j + 1]) * invg) * E2[i + 1];
            }
            float w = (float)(i - j);
            w = w * w * wscale;                    // omega(i,j) = (i-j)^2/(T-1)^2
            sumE  += e;
            sumEw += e * w;
        }
        Ec[i] = e;
        __syncthreads();
    }

    // ---- block reductions ----
    red[i] = sumE;
    __syncthreads();
    for (int s = 128; s > 0; s >>= 1) {
        if (i < s) red[i] += red[i + s];
        __syncthreads();
    }
    float totE = red[0];
    __syncthreads();
    red[i] = sumEw;
    __syncthreads();
    for (int s = 128; s > 0; s >>= 1) {
        if (i < s) red[i] += red[i + s];
        __syncthreads();
    }
    if (i == 0) {
        partials[b]          = R[(size_t)NT * RP + NT];  // R_{T,T}
        partials[NB + b]     = red[0];                   // sum(E * omega)
        partials[2 * NB + b] = totE;                     // sum(E)
    }
}

// ---------------------------------------------------------------------------
// Kernel 3: combine per-batch partials into the scalar loss.
// alignment = E/T (vjp of R/T); the 1/T cancels in the temporal ratio except
// inside max(., eps), which we restore explicitly.
// ---------------------------------------------------------------------------
__global__ void dilate_final(const float* __restrict__ partials,
                             float* __restrict__ out) {
    __shared__ float red[NB];
    int t = threadIdx.x;
    float rfin = partials[t];
    float sEw  = partials[NB + t];
    float sE   = partials[2 * NB + t];
    float shape = rfin / (float)NT;
    float tv    = (sEw / (float)NT) / fmaxf(sE / (float)NT, 1e-8f);
    red[t] = 0.5f * shape + 0.5f * tv;       // alpha = 0.5
    __syncthreads();
    for (int s = NB / 2; s > 0; s >>= 1) {
        if (t < s) red[t] += red[t + s];
        __syncthreads();
    }
    if (t == 0) out[0] = red[0] / (float)NB;
}

// ---------------------------------------------------------------------------
extern "C" void kernel_launch(void* const* d_in, const int* in_sizes, int n_in,
                              void* d_out, int out_size, void* d_ws, size_t ws_size,
                              hipStream_t stream) {
    const float* preds   = (const float*)d_in[0];
    const float* targets = (const float*)d_in[1];
    float* ws = (float*)d_ws;

    // workspace layout (floats): pn | tn | D | partials   (~33.8 MB total)
    float* pn       = ws;
    float* tn       = ws + (size_t)NB * NT;
    float* Dg       = ws + (size_t)2 * NB * NT;
    float* partials = Dg + (size_t)NB * NT * NT;

    dilate_norms<<<(NB * NT) / 256, 256, 0, stream>>>(preds, targets, pn, tn);

    // 32768 tiles, 8 waves (one tile each) per 256-thread block
    dilate_dist_wmma<<<(NB * 256) / 8, 256, 0, stream>>>(preds, targets, pn, tn, Dg);

    size_t shmem = (size_t)(RROWS * RP + 3 * EBUF + 256) * sizeof(float); // ~269 KB < 320 KB
    dilate_softdtw<<<NB, 256, shmem, stream>>>(Dg, partials);

    dilate_final<<<1, NB, 0, stream>>>(partials, (float*)d_out);
}